// LinearCRF_25168508355383
// MI455X (gfx1250) — compile-verified
//
#include <hip/hip_runtime.h>
#include <cstdint>

// ---------------------------------------------------------------------------
// Viterbi decode for a linear-chain CRF on gfx1250 (MI455X).
//   feats: (B, S, T) f32, mask: (B, S) bool(u8), trans: (T, T) f32
//   out:   (B, S) tag indices (written as float)
// One workgroup (128 threads, 4 wave32) per batch element; CDNA5-only LDS
// budget (~134 KB of the 320 KB WGP LDS):
//   [0,      66K)  transitions TRANSPOSED with pitch 132 floats:
//                  trans_t[j*132 + i] = trans[i*128 + j]  -> thread j reads
//                  its column as contiguous float4 (ds_load_b128, full LDS BW)
//   [66K,   130K)  backpointer history, uint8[S][T]
//   [130K, +1K)    ping/pong partition vectors (2 x 128 f32) + scratch
// ---------------------------------------------------------------------------

constexpr int kT   = 128;       // tag_size
constexpr int kS   = 512;       // sequence length
constexpr int kTP  = 132;       // padded LDS pitch (floats) -> 2-way max conflicts
constexpr int kStart = kT - 3;  // 125
constexpr int kStop  = kT - 2;  // 126

constexpr int kOffTrans = 0;
constexpr int kOffBp    = kT * kTP * 4;            // 67584
constexpr int kOffPartA = kOffBp + kS * kT;        // 133120 (16B aligned)
constexpr int kOffPartB = kOffPartA + kT * 4;
constexpr int kOffScr   = kOffPartB + kT * 4;
constexpr int kSmemSize = kOffScr + 64;            // ~134 KB

typedef unsigned int u32x4 __attribute__((ext_vector_type(4)));
typedef int          i32x4 __attribute__((ext_vector_type(4)));
typedef int          i32x8 __attribute__((ext_vector_type(8)));
typedef float        f32x4 __attribute__((ext_vector_type(4)));

#if defined(__has_builtin)
#if __has_builtin(__builtin_amdgcn_tensor_load_to_lds) && \
    __has_builtin(__builtin_amdgcn_s_wait_tensorcnt)
#define VIT_HAVE_TDM 1
#endif
#endif

__device__ __forceinline__ void tdm_load_trans_to_lds(const float* gsrc,
                                                      void* lds_dst) {
#ifdef VIT_HAVE_TDM
  // Tensor DMA Descriptor (D#) per CDNA5 ISA ch.8:
  // 2D tensor, data_size=4B, tensor 128x128, tile 128x128, stride0=128.
  uint64_t ga = (uint64_t)(uintptr_t)gsrc;
  uint32_t la = (uint32_t)(uintptr_t)lds_dst;  // low 32b = LDS offset

  u32x4 g0;
  g0[0] = 1u;                                        // count=1, user mode
  g0[1] = la;                                        // lds_addr
  g0[2] = (uint32_t)(ga & 0xFFFFFFFFu);              // global_addr[31:0]
  g0[3] = (uint32_t)((ga >> 32) & 0x1FFFFFFu)        // global_addr[56:32]
        | (2u << 30);                                // type=2 ("image")

  i32x8 g1;
  g1[0] = (int)(2u << 16);          // workgroup_mask=0, data_size=2 (4 bytes)
  g1[1] = (int)(128u << 16);        // tensor_dim0[15:0]=128 at bits[31:16]
  g1[2] = (int)(128u << 16);        // tensor_dim0 hi=0 | tensor_dim1 lo=128
  g1[3] = (int)(128u << 16);        // tensor_dim1 hi=0 | tile_dim0=128
  g1[4] = (int)(128u);              // tile_dim1=128, tile_dim2=0
  g1[5] = (int)(128u);              // tensor_dim0_stride[31:0] = 128
  g1[6] = 0;
  g1[7] = 0;

  i32x4 z4 = {0, 0, 0, 0};
#if __has_include(<hip/amd_detail/amd_gfx1250_TDM.h>)
  i32x8 z8 = {0, 0, 0, 0, 0, 0, 0, 0};
  __builtin_amdgcn_tensor_load_to_lds(g0, g1, z4, z4, z8, 0);   // 6-arg lane
#else
  __builtin_amdgcn_tensor_load_to_lds(g0, g1, z4, z4, 0);       // 5-arg lane
#endif
  __builtin_amdgcn_s_wait_tensorcnt(0);
#else
  (void)gsrc; (void)lds_dst;
#endif
}

__global__ __launch_bounds__(kT)
void viterbi_crf_kernel(const float* __restrict__ feats,
                        const unsigned char* __restrict__ mask,
                        const float* __restrict__ trans,
                        float* __restrict__ out) {
  extern __shared__ char smem[];
  float*         trans_t = (float*)(smem + kOffTrans);   // kT rows, pitch kTP
  unsigned char* bp_s    = (unsigned char*)(smem + kOffBp);
  float*         partA   = (float*)(smem + kOffPartA);
  float*         partB   = (float*)(smem + kOffPartB);
  int*           scr     = (int*)(smem + kOffScr);       // [0]=len, [1]=ptr

  const int j = threadIdx.x;         // tag column owned by this thread
  const int b = blockIdx.x;          // batch element
  const size_t featBase = (size_t)b * kS * kT;

  // ---- TDM async DMA of the 64KB transition tile into LDS (prefetch) ----
  if (threadIdx.x < 32) {            // one wave32 issues; EXEC-uniform branch
    tdm_load_trans_to_lds(trans, trans_t);
  }
  __syncthreads();
  if (j == 0) scr[0] = 0;
  // Authoritative transposed fill: coalesced global reads, LDS scatter.
  for (int m = j; m < kT * kT; m += kT) {
    int r = m >> 7, c = m & (kT - 1);
    trans_t[c * kTP + r] = trans[m];
  }
  __syncthreads();

  // ---- Sequence length from prefix mask (LDS ds_add reduction) ----
  int lsum = 0;
  for (int k = j; k < kS; k += kT) lsum += (int)mask[(size_t)b * kS + k];
  if (lsum) atomicAdd(scr, lsum);

  // ---- part0 = emit[0] + trans[START, :]  (same add order as reference) ----
  partA[j] = feats[featBase + j] + trans_t[j * kTP + kStart];
  __syncthreads();

  const int len = scr[0];
  const int last_pos = len - 1;

  // ---- Forward max-plus recurrence with backpointers ----
  const float* trow = trans_t + j * kTP;   // this thread's transition column
  float* cur = partA;
  float* nxt = partB;
  for (int t = 1; t <= last_pos; ++t) {
    const float* erow = feats + featBase + (size_t)t * kT;
    if (t + 4 <= last_pos)
      __builtin_prefetch(feats + featBase + (size_t)(t + 4) * kT + j, 0, 1);
    const float e = erow[j];
    const f32x4 ev = {e, e, e, e};
    const f32x4* curv = (const f32x4*)cur;

    // Four independent first-argmax chains over contiguous quarters
    // [0,32) [32,64) [64,96) [96,128): ILP for the cmp/cndmask chain while
    // preserving jnp.argmax first-max semantics exactly.
    float b0 = -__builtin_inff(), b1 = b0, b2 = b0, b3 = b0;
    int   a0 = 0, a1 = 0, a2 = 0, a3 = 0;
#pragma unroll
    for (int k = 0; k < 8; ++k) {
      const int i0 = 4 * k;
      f32x4 t0 = *(const f32x4*)(trow + i0);
      f32x4 t1 = *(const f32x4*)(trow + 32 + i0);
      f32x4 t2 = *(const f32x4*)(trow + 64 + i0);
      f32x4 t3 = *(const f32x4*)(trow + 96 + i0);
      f32x4 p0 = curv[k], p1 = curv[8 + k], p2 = curv[16 + k], p3 = curv[24 + k];
      // (part + trans) + e, elementwise -> v_pk_add_f32-friendly, bit-exact
      f32x4 c0 = (p0 + t0) + ev;
      f32x4 c1 = (p1 + t1) + ev;
      f32x4 c2 = (p2 + t2) + ev;
      f32x4 c3 = (p3 + t3) + ev;
      if (c0.x > b0) { b0 = c0.x; a0 = i0; }
      if (c1.x > b1) { b1 = c1.x; a1 = i0; }
      if (c2.x > b2) { b2 = c2.x; a2 = i0; }
      if (c3.x > b3) { b3 = c3.x; a3 = i0; }
      if (c0.y > b0) { b0 = c0.y; a0 = i0 + 1; }
      if (c1.y > b1) { b1 = c1.y; a1 = i0 + 1; }
      if (c2.y > b2) { b2 = c2.y; a2 = i0 + 1; }
      if (c3.y > b3) { b3 = c3.y; a3 = i0 + 1; }
      if (c0.z > b0) { b0 = c0.z; a0 = i0 + 2; }
      if (c1.z > b1) { b1 = c1.z; a1 = i0 + 2; }
      if (c2.z > b2) { b2 = c2.z; a2 = i0 + 2; }
      if (c3.z > b3) { b3 = c3.z; a3 = i0 + 2; }
      if (c0.w > b0) { b0 = c0.w; a0 = i0 + 3; }
      if (c1.w > b1) { b1 = c1.w; a1 = i0 + 3; }
      if (c2.w > b2) { b2 = c2.w; a2 = i0 + 3; }
      if (c3.w > b3) { b3 = c3.w; a3 = i0 + 3; }
    }
    // Merge quarters in ascending order, strict '>' => ties keep lower index.
    float best = b0; int arg = a0;
    if (b1 > best) { best = b1; arg = 32 + a1; }
    if (b2 > best) { best = b2; arg = 64 + a2; }
    if (b3 > best) { best = b3; arg = 96 + a3; }

    nxt[j] = best;
    bp_s[(t - 1) * kT + j] = (unsigned char)arg;
    __syncthreads();
    float* tmp = cur; cur = nxt; nxt = tmp;
  }

  // ---- Final pointer: first-argmax_i( part_last[i] + trans[i][STOP] ) ----
  if (j == 0) {
    const float* stoprow = trans_t + kStop * kTP;   // trans[i][STOP], i contiguous
    float bestv = -__builtin_inff();
    int   bi = 0;
    for (int i = 0; i < kT; ++i) {
      float c = cur[i] + stoprow[i];
      if (c > bestv) { bestv = c; bi = i; }
    }
    scr[1] = bi;
  }
  __syncthreads();
  const int pointer = scr[1];

  // ---- Emit decode row (all S positions written) ----
  float* orow = out + (size_t)b * kS;
  for (int t = j; t < kS; t += kT) {
    if (t > last_pos && t < kS - 1) orow[t] = 0.0f;   // masked-region zeros
  }
  if (j == 0) {
    orow[kS - 1]   = (float)pointer;
    orow[last_pos] = (float)pointer;
    // Traceback: p_t = bp_step[t+1][p_{t+1}], stored at bp_s[t*kT + .]
    int ptr = pointer;
    for (int t = last_pos - 1; t >= 0; --t) {
      ptr = (int)bp_s[t * kT + ptr];
      orow[t] = (float)ptr;
    }
  }
}

extern "C" void kernel_launch(void* const* d_in, const int* in_sizes, int n_in,
                              void* d_out, int out_size, void* d_ws, size_t ws_size,
                              hipStream_t stream) {
  (void)n_in; (void)out_size; (void)d_ws; (void)ws_size;
  const float*         feats = (const float*)d_in[0];
  const unsigned char* mask  = (const unsigned char*)d_in[1];  // jnp bool_ = 1 byte
  const float*         trans = (const float*)d_in[2];
  float*               out   = (float*)d_out;

  const int B = in_sizes[0] / (kS * kT);   // 256
  viterbi_crf_kernel<<<dim3(B), dim3(kT), (size_t)kSmemSize, stream>>>(
      feats, mask, trans, out);
}